// PCENExtractor_86835648790683
// MI455X (gfx1250) — compile-verified
//
#include <hip/hip_runtime.h>
#include <math.h>

// ---------------- problem constants ----------------
#define SRATE     22050
#define NFFT      1024
#define HOP       256
#define NMELS     128
#define NFREQS    513
#define FPAD      528              // 513 padded to 33 tiles of 16
#define BATCH     16
#define NSAMP     661504
#define NFRAMES   2585             // 1 + NSAMP/HOP
#define TTILES    162              // ceil(2585/16)
#define SPAN      (15*HOP + NFFT)  // 4864 samples per 16-frame tile
#define NCH       (BATCH*NMELS)    // 2048 channels for PCEN

typedef __attribute__((ext_vector_type(2))) float v2f;
typedef __attribute__((ext_vector_type(8))) float v8f;

// ---------------- table builders ----------------
// DFT matrices, Hann window folded in. Packed so that K-pairs (k0, k0+1) are
// adjacent: dft[(k>>1)*(FPAD*2) + f*2 + (k&1)]  -> one float2 per B fragment.
__global__ void init_dft(float* __restrict__ dftCos, float* __restrict__ dftSin) {
    int idx = blockIdx.x * blockDim.x + threadIdx.x;
    if (idx >= NFFT * FPAD) return;
    int k = idx / FPAD;
    int f = idx % FPAD;
    float c = 0.0f, s = 0.0f;
    if (f < NFREQS) {
        float win = 0.5f - 0.5f * cosf(6.28318530717958647692f * (float)k / (float)NFFT);
        int r = (k * f) & (NFFT - 1);                  // exact phase reduction
        float ang = (float)r * (6.28318530717958647692f / (float)NFFT);
        c = win * cosf(ang);
        s = win * sinf(ang);                           // sign irrelevant for |X|
    }
    int off = (k >> 1) * (FPAD * 2) + f * 2 + (k & 1);
    dftCos[off] = c;
    dftSin[off] = s;
}

__device__ __forceinline__ float hz_to_mel(float f) { return 2595.0f * log10f(1.0f + f / 700.0f); }
__device__ __forceinline__ float mel_to_hz(float m) { return 700.0f * (exp10f(m / 2595.0f) - 1.0f); }

// HTK mel filterbank, packed like the DFT tables: mel[(k>>1)*(NMELS*2) + m*2 + (k&1)]
__global__ void init_mel(float* __restrict__ melFB) {
    int idx = blockIdx.x * blockDim.x + threadIdx.x;
    if (idx >= FPAD * NMELS) return;
    int k = idx / NMELS;   // freq bin (padded)
    int m = idx % NMELS;
    float v = 0.0f;
    if (k < NFREQS) {
        float m_lo = hz_to_mel(50.0f);
        float m_hi = hz_to_mel((float)SRATE * 0.5f);
        float step = (m_hi - m_lo) / (float)(NMELS + 1);
        float f0 = mel_to_hz(m_lo + step * (float)m);
        float f1 = mel_to_hz(m_lo + step * (float)(m + 1));
        float f2 = mel_to_hz(m_lo + step * (float)(m + 2));
        float freq = (float)k * ((float)SRATE * 0.5f / (float)(NFFT / 2));
        float down = (freq - f0) / (f1 - f0);
        float up   = (f2 - freq) / (f2 - f1);
        v = fmaxf(0.0f, fminf(down, up));
    }
    melFB[(k >> 1) * (NMELS * 2) + m * 2 + (k & 1)] = v;
}

// ---------------- fused DFT-magnitude + mel GEMM (f32 WMMA) ----------------
__global__ __launch_bounds__(128)
void dft_mel_kernel(const float* __restrict__ x,
                    const float* __restrict__ dftCos,
                    const float* __restrict__ dftSin,
                    const float* __restrict__ melFB,
                    float* __restrict__ melBuf)   // [T][NCH]
{
    __shared__ float span[SPAN];
    __shared__ float mag[16 * FPAD];

    const int tile = blockIdx.x;        // frame tile
    const int b    = blockIdx.y;        // batch
    const int tid  = threadIdx.x;
    const int ln   = tid & 31;          // lane in wave32
    const int wv   = tid >> 5;          // wave id 0..3
    const int t0   = tile * 16;

    // --- cooperative load of the overlapping sample span, reflect-padded ---
    const float* xb = x + (size_t)b * NSAMP;
    const long base = (long)t0 * HOP - (NFFT / 2);
    for (int i = tid; i < SPAN; i += 128) {
        long p = base + i;
        if (p < 0) p = -p;
        if (p >= NSAMP) p = 2L * NSAMP - 2 - p;
        if (p < 0) p = 0;
        if (p >= NSAMP) p = NSAMP - 1;
        span[i] = xb[p];
    }
    __syncthreads();

    const int mrow  = ln & 15;           // A-matrix row (frame within tile)
    const int kh    = (ln >> 4) << 1;    // K offset 0 / 2 per 32b 16x4 layout
    const int ncol  = ln & 15;           // B/D column within a 16-wide tile
    const int mbase = (ln >> 4) * 8;     // D-matrix row base per lane half

    // ===== stage 1: |DFT| via V_WMMA_F32_16X16X4_F32, K = 1024 =====
    for (int nt = wv; nt < FPAD / 16; nt += 4) {
        v8f accRe = {0.f, 0.f, 0.f, 0.f, 0.f, 0.f, 0.f, 0.f};
        v8f accIm = {0.f, 0.f, 0.f, 0.f, 0.f, 0.f, 0.f, 0.f};
        const int n = nt * 16 + ncol;
        #pragma unroll 4
        for (int kb = 0; kb < NFFT; kb += 4) {
            const int k0 = kb + kh;
            v2f a  = *(const v2f*)&span[mrow * HOP + k0];
            v2f bc = *(const v2f*)&dftCos[(k0 >> 1) * (FPAD * 2) + n * 2];
            v2f bs = *(const v2f*)&dftSin[(k0 >> 1) * (FPAD * 2) + n * 2];
            accRe = __builtin_amdgcn_wmma_f32_16x16x4_f32(false, a, false, bc,
                                                          (short)0, accRe, false, false);
            accIm = __builtin_amdgcn_wmma_f32_16x16x4_f32(false, a, false, bs,
                                                          (short)0, accIm, false, false);
        }
        #pragma unroll
        for (int v = 0; v < 8; ++v) {
            float re = accRe[v], im = accIm[v];
            mag[(mbase + v) * FPAD + n] = sqrtf(re * re + im * im);
        }
    }
    __syncthreads();

    // ===== stage 2: mel projection via V_WMMA_F32_16X16X4_F32, K = 528 =====
    for (int nt = wv; nt < NMELS / 16; nt += 4) {
        v8f acc = {0.f, 0.f, 0.f, 0.f, 0.f, 0.f, 0.f, 0.f};
        const int n = nt * 16 + ncol;    // mel index
        #pragma unroll 4
        for (int kb = 0; kb < FPAD; kb += 4) {
            const int k0 = kb + kh;
            v2f a  = *(const v2f*)&mag[mrow * FPAD + k0];
            v2f bm = *(const v2f*)&melFB[(k0 >> 1) * (NMELS * 2) + n * 2];
            acc = __builtin_amdgcn_wmma_f32_16x16x4_f32(false, a, false, bm,
                                                        (short)0, acc, false, false);
        }
        #pragma unroll
        for (int v = 0; v < 8; ++v) {
            int t = t0 + mbase + v;
            if (t < NFRAMES)
                melBuf[(size_t)t * NCH + b * NMELS + n] = acc[v];
        }
    }
}

// ---------------- PCEN: per-channel sequential EMA + pointwise ----------------
__global__ void pcen_kernel(const float* __restrict__ melBuf,   // [T][NCH]
                            float* __restrict__ out)            // [B][1][NMELS][T]
{
    int ch = blockIdx.x * blockDim.x + threadIdx.x;
    if (ch >= NCH) return;
    int b = ch / NMELS, m = ch % NMELS;
    const float s  = 1.0f - expf(-((float)HOP / (float)SRATE) / 0.06f);
    const float da = powf(2.0f, 0.98f);
    float* o = out + ((size_t)b * NMELS + m) * NFRAMES;
    float M = 0.0f;
    for (int t = 0; t < NFRAMES; ++t) {
        float e = melBuf[(size_t)t * NCH + ch];   // channel-contiguous -> coalesced
        M = (1.0f - s) * M + s * e;
        float smooth = sqrtf(M + 1e-6f);          // (M+eps)^0.5
        o[t] = powf(e / smooth + 2.0f, 0.98f) - da;
    }
}

// ---------------- entry ----------------
extern "C" void kernel_launch(void* const* d_in, const int* in_sizes, int n_in,
                              void* d_out, int out_size, void* d_ws, size_t ws_size,
                              hipStream_t stream) {
    const float* x = (const float*)d_in[0];
    float* ws = (float*)d_ws;

    float* dftCos = ws;                                   // 1024*528 f32
    float* dftSin = dftCos + (size_t)NFFT * FPAD;         // 1024*528 f32
    float* melFB  = dftSin + (size_t)NFFT * FPAD;         //  528*128 f32
    float* melBuf = melFB  + (size_t)FPAD * NMELS;        // 2592*2048 f32 region (2585 used)

    int n1 = NFFT * FPAD;
    init_dft<<<(n1 + 255) / 256, 256, 0, stream>>>(dftCos, dftSin);
    int n2 = FPAD * NMELS;
    init_mel<<<(n2 + 255) / 256, 256, 0, stream>>>(melFB);

    dim3 grid(TTILES, BATCH);
    dft_mel_kernel<<<grid, 128, 0, stream>>>(x, dftCos, dftSin, melFB, melBuf);

    pcen_kernel<<<(NCH + 127) / 128, 128, 0, stream>>>(melBuf, (float*)d_out);
}